// MatrixTreeMarkovICL_14027363188797
// MI455X (gfx1250) — compile-verified
//
#include <hip/hip_runtime.h>
#include <hip/hip_bf16.h>
#include <math.h>

// ---------------------------------------------------------------------------
// Problem constants (from reference)
// ---------------------------------------------------------------------------
#define BATCH   4096
#define NN      32          // N_NODES
#define ZFULL   272         // (N_CTX+1)*Z_DIM
#define NCTX    16
#define NMAT    1024        // NN*NN
#define MINOR   31          // NN-1

typedef __attribute__((ext_vector_type(2))) float v2f;
typedef __attribute__((ext_vector_type(8))) float v8f;

// ---------------------------------------------------------------------------
// Kernel 1: rate_mod GEMM (4096x272 @ 272x1024) with fp32 WMMA, fused
//           clip/exp epilogue -> rate matrices R[b][i][j] (diag left 0).
//   A = z_flat        (row-major, stride ZFULL)
//   B = K_params as (1024 x 272) row-major; output col n = i*32+j uses row n.
// One wave computes one 16x16 output tile; 68 k-steps of V_WMMA_F32_16X16X4.
// ---------------------------------------------------------------------------
__global__ __launch_bounds__(256)
void rate_gemm_kernel(const float* __restrict__ z,      // [BATCH][ZFULL]
                      const float* __restrict__ kmat,   // [NMAT][ZFULL]
                      const float* __restrict__ base,   // [NN][NN] = [NMAT]
                      float* __restrict__ R)            // [BATCH][NMAT]
{
    const int lane   = threadIdx.x & 31;
    const int wave   = threadIdx.x >> 5;
    const int tile   = blockIdx.x * 8 + wave;          // 16384 tiles total
    const int tilesN = NMAT / 16;                      // 64
    const int n0     = (tile % tilesN) * 16;
    const int m0     = (tile / tilesN) * 16;

    const int row  = lane & 15;          // M (for A) / N (for B)
    const int koff = (lane >> 4) << 1;   // 0 or 2

    const float* __restrict__ zrow = z    + (size_t)(m0 + row) * ZFULL + koff;
    const float* __restrict__ krow = kmat + (size_t)(n0 + row) * ZFULL + koff;

    v8f c = {};
    #pragma unroll 4
    for (int k0 = 0; k0 < ZFULL; k0 += 4) {
        v2f a, b;
        a.x = zrow[k0];  a.y = zrow[k0 + 1];
        b.x = krow[k0];  b.y = krow[k0 + 1];
        // D = A(16x4) * B(4x16) + C   (fp32 WMMA, wave32)
        c = __builtin_amdgcn_wmma_f32_16x16x4_f32(
                /*neg_a=*/false, a, /*neg_b=*/false, b,
                /*c_mod=*/(short)0, c, /*reuse_a=*/false, /*reuse_b=*/false);
    }

    // Epilogue: lane owns column n = n0 + (lane&15); rows m0+v+8*(lane>>4).
    const int   n      = n0 + (lane & 15);
    const bool  isdiag = ((n >> 5) == (n & 31));       // i == j
    const float bl     = base[n];
    const int   mbase  = m0 + ((lane >> 4) << 3);

    #pragma unroll
    for (int v = 0; v < 8; ++v) {
        float lr   = fminf(fmaxf(bl + c[v], -15.0f), 3.0f);
        float rate = isdiag ? 0.0f : __expf(lr);
        R[(size_t)(mbase + v) * NMAT + n] = rate;
    }
}

// ---------------------------------------------------------------------------
// Kernel 2: per batch item, set diagonal, compute 32 minor determinants
//           (Gaussian elimination, one wave per 31x31 minor, lanes = columns),
//           matrix-tree normalization, gamma contraction -> scalar out.
// Block = 256 threads = 8 waves; wave w handles roots w, w+8, w+16, w+24.
// ---------------------------------------------------------------------------
__global__ __launch_bounds__(256)
void tree_det_kernel(const float* __restrict__ R,       // [BATCH][NMAT]
                     const float* __restrict__ Bp,      // [NN][1][NCTX]
                     const float* __restrict__ labels,  // [BATCH][NCTX]
                     float* __restrict__ out)           // [BATCH]
{
    __shared__ float Rl[NN][NN];          // 4 KB   rate matrix
    __shared__ float Mw[8][MINOR][32];    // 31 KB  per-wave minor scratch
    __shared__ float dets[NN];
    __shared__ int   badflag;

    const int b    = blockIdx.x;
    const int tid  = threadIdx.x;
    const int wave = tid >> 5;
    const int lane = tid & 31;

    if (tid == 0) badflag = 0;

    // Load 32x32 rate matrix (diag currently 0).
    #pragma unroll
    for (int e = 0; e < 4; ++e) {
        int idx = tid + e * 256;
        Rl[idx >> 5][idx & 31] = R[(size_t)b * NMAT + idx];
    }
    __syncthreads();

    // diag[j] = -sum_i rates[i][j]  (column sums; diag entry is 0 so harmless)
    if (tid < NN) {
        float s = 0.0f;
        #pragma unroll
        for (int i = 0; i < NN; ++i) s += Rl[i][tid];
        Rl[tid][tid] = -s;
    }
    __syncthreads();

    // Each wave: Gaussian elimination on its minors. volatile => DS accesses
    // stay ordered; cross-lane LDS dependences within a lockstep wave are
    // safe because same-wave DS ops complete in order.
    volatile float* M = (volatile float*)&Mw[wave][0][0];   // row stride 32

    for (int rt = wave; rt < NN; rt += 8) {
        // Copy minor (delete row rt, col rt).
        if (lane < MINOR) {
            const int jc = lane + (lane >= rt);
            #pragma unroll 4
            for (int a = 0; a < MINOR; ++a) {
                const int ia = a + (a >= rt);
                M[a * 32 + lane] = Rl[ia][jc];
            }
        }

        float det = 1.0f;
        for (int k = 0; k < MINOR; ++k) {
            const float piv = M[k * 32 + k];        // broadcast read
            det *= piv;
            const float pr  = (lane < MINOR) ? M[k * 32 + lane] : 0.0f;
            const float inv = 1.0f / piv;
            for (int r = k + 1; r < MINOR; ++r) {
                const float f = M[r * 32 + k] * inv;   // broadcast read
                if (lane < MINOR)
                    M[r * 32 + lane] = M[r * 32 + lane] - f * pr;
            }
        }

        if (lane == 0) {
            if (!__builtin_isfinite(det)) atomicOr(&badflag, 1);
            dets[rt] = fminf(fmaxf(fabsf(det), 1e-10f), 1e10f);
        }
    }
    __syncthreads();

    // Wave 0: Z, p, gamma, output scalar.
    if (tid < NN) {
        const int i = tid;
        float d = dets[i];
        float Zs = d;
        #pragma unroll
        for (int off = 16; off > 0; off >>= 1) Zs += __shfl_xor(Zs, off, 32);
        Zs = fmaxf(Zs, 1e-8f);
        float p = badflag ? (1.0f / (float)NN) : (d / Zs);

        float g = 0.0f;
        #pragma unroll
        for (int dd = 0; dd < NCTX; ++dd)
            g += Bp[i * NCTX + dd] * labels[(size_t)b * NCTX + dd];

        float o = p * g;
        #pragma unroll
        for (int off = 16; off > 0; off >>= 1) o += __shfl_xor(o, off, 32);
        if (tid == 0) out[b] = o;
    }
}

// ---------------------------------------------------------------------------
// Launch
// ---------------------------------------------------------------------------
extern "C" void kernel_launch(void* const* d_in, const int* in_sizes, int n_in,
                              void* d_out, int out_size, void* d_ws, size_t ws_size,
                              hipStream_t stream) {
    const float* z      = (const float*)d_in[0];  // [4096][17][16]
    const float* labels = (const float*)d_in[1];  // [4096][16]
    const float* kmat   = (const float*)d_in[2];  // [32][32][272]
    const float* Bp     = (const float*)d_in[3];  // [32][1][16]
    const float* base   = (const float*)d_in[4];  // [32][32]
    float*       out    = (float*)d_out;          // [4096]
    float*       R      = (float*)d_ws;           // [4096][1024] = 16.8 MB

    // 16384 output tiles of 16x16, 8 waves per block -> 2048 blocks
    rate_gemm_kernel<<<2048, 256, 0, stream>>>(z, kmat, base, R);
    tree_det_kernel<<<BATCH, 256, 0, stream>>>(R, Bp, labels, out);
}